// interactionModule_77936476554069
// MI455X (gfx1250) — compile-verified
//
#include <hip/hip_runtime.h>
#include <hip/hip_bf16.h>
#include <stdint.h>

#define GAMMA 0.1f
#define EPSF  1e-12f

#define TPB   256          // threads per block (8 wave32)
#define EPT   4            // edges per thread per tile (one b128 = 4 int32)
#define TILE  (TPB * EPT)  // 1024 edges per tile

// ---------------------------------------------------------------------------
// CDNA5 async global -> LDS copy (per-lane 16B), tracked with ASYNCcnt.
// vdst = VGPR holding LDS byte offset, vaddr = 64-bit global address pair.
// ---------------------------------------------------------------------------
__device__ __forceinline__ void async_copy_b128(unsigned lds_off, const void* gptr)
{
    asm volatile("global_load_async_to_lds_b128 %0, %1, off"
                 :: "v"(lds_off), "v"(gptr)
                 : "memory");
}

template <int N>
__device__ __forceinline__ void wait_asynccnt()
{
    asm volatile("s_wait_asynccnt %0" :: "n"(N) : "memory");
}

__device__ __forceinline__ void wait_dscnt0()
{
    asm volatile("s_wait_dscnt 0" ::: "memory");
}

// ---------------------------------------------------------------------------
// out = -gamma * v   (also re-initializes d_out deterministically every call)
// ---------------------------------------------------------------------------
__global__ void node_damp_kernel(const float2* __restrict__ v,
                                 float2* __restrict__ out, int n)
{
    int i = blockIdx.x * blockDim.x + threadIdx.x;
    int stride = gridDim.x * blockDim.x;
    for (; i < n; i += stride) {
        float2 vv = v[i];
        float2 o;
        o.x = -GAMMA * vv.x;
        o.y = -GAMMA * vv.y;
        out[i] = o;
    }
}

// ---------------------------------------------------------------------------
// Edge kernel: stream src/dst indices through LDS with double-buffered async
// copies; gather x (L2-resident); compute zero-potential message; atomic
// scatter to out. Each lane reads only the LDS bytes its own async op wrote,
// so per-wave s_wait_asynccnt is sufficient (no workgroup barriers).
// ---------------------------------------------------------------------------
__global__ __launch_bounds__(TPB)
void edge_force_kernel(const float2* __restrict__ x,
                       const int*    __restrict__ src,
                       const int*    __restrict__ dst,
                       float*        __restrict__ out,
                       int E, int numTiles)
{
    __shared__ alignas(16) int s_idx[2][2][TILE];   // [buf][src/dst][edge-in-tile]

    const int tid  = threadIdx.x;
    const int slot = tid * EPT;

    // LDS byte offsets of this lane's 16B staging slots (low 32 bits of the
    // generic pointer == hardware LDS address).
    unsigned lsrc[2], ldst[2];
    lsrc[0] = (unsigned)(uintptr_t)&s_idx[0][0][slot];
    ldst[0] = (unsigned)(uintptr_t)&s_idx[0][1][slot];
    lsrc[1] = (unsigned)(uintptr_t)&s_idx[1][0][slot];
    ldst[1] = (unsigned)(uintptr_t)&s_idx[1][1][slot];

    auto issue = [&](int tile, int buf) {
        long base = (long)tile * TILE + slot;
        long bmax = (long)E - EPT;          // host guarantees E >= EPT
        if (base > bmax) base = bmax;       // clamp tail (extra data ignored)
        async_copy_b128(lsrc[buf], src + base);
        async_copy_b128(ldst[buf], dst + base);
    };

    int tile = blockIdx.x;
    if (tile >= numTiles) return;

    issue(tile, 0);                         // prologue: ASYNCcnt = 2
    int buf = 0;

    for (; tile < numTiles; tile += gridDim.x) {
        const int nextTile = tile + gridDim.x;
        if (nextTile < numTiles) {
            wait_dscnt0();                  // WAR: prior reads of buf^1 retired
            issue(nextTile, buf ^ 1);       // ASYNCcnt = 4
            wait_asynccnt<2>();             // current tile's 2 loads complete
        } else {
            wait_asynccnt<0>();
        }

        const long ebase = (long)tile * TILE + slot;
#pragma unroll
        for (int k = 0; k < EPT; ++k) {
            const long e = ebase + k;
            if (e < E) {
                const int i = s_idx[buf][0][slot + k];
                const int j = s_idx[buf][1][slot + k];
                const float2 xi = x[i];
                const float2 xj = x[j];
                const float drx = xj.x - xi.x;
                const float dry = xj.y - xi.y;
                const float r    = __builtin_amdgcn_sqrtf(drx * drx + dry * dry);
                const float rinv = __builtin_amdgcn_rcpf(fmaxf(r, EPSF));
                const float force = r * 0.0f;        // zeroPotential.force(r)
                const float mx = force * (drx * rinv);
                const float my = force * (dry * rinv);
                unsafeAtomicAdd(&out[2 * (long)j + 0], mx);  // global_atomic_add_f32
                unsafeAtomicAdd(&out[2 * (long)j + 1], my);
            }
        }
        buf ^= 1;
    }
}

// ---------------------------------------------------------------------------
extern "C" void kernel_launch(void* const* d_in, const int* in_sizes, int n_in,
                              void* d_out, int out_size, void* d_ws, size_t ws_size,
                              hipStream_t stream)
{
    const float2* x   = (const float2*)d_in[0];
    const float2* v   = (const float2*)d_in[1];
    const int*    src = (const int*)d_in[2];
    const int*    dst = (const int*)d_in[3];
    float*        out = (float*)d_out;

    const int N = in_sizes[0] / 2;
    const int E = in_sizes[2];

    // 1) out = -gamma * v  (deterministic full overwrite of d_out)
    {
        int blocks = (N + TPB - 1) / TPB;
        if (blocks > 2048) blocks = 2048;
        if (blocks > 0)
            node_damp_kernel<<<blocks, TPB, 0, stream>>>(v, (float2*)out, N);
    }

    // 2) edge messages (exactly zero-valued, computed faithfully) scattered in
    if (E >= EPT) {
        const int numTiles = (E + TILE - 1) / TILE;
        int blocks = numTiles < 2048 ? numTiles : 2048;
        edge_force_kernel<<<blocks, TPB, 0, stream>>>(x, src, dst, out, E, numTiles);
    }
}